// MaskedSelfAttention_78451872629488
// MI455X (gfx1250) — compile-verified
//
#include <hip/hip_runtime.h>
#include <hip/hip_bf16.h>

typedef __attribute__((ext_vector_type(2))) float v2f;
typedef __attribute__((ext_vector_type(8))) float v8f;

#define EMBED   768
#define HEADS   12
#define HDIM    64
#define IMG     196
#define TXT     1852
#define SEQ     2048
#define BATCHN  4
#define NEGV    (-1e9f)
#define KVPAD   68      // LDS row stride (floats) -> conflict-free fragment reads

// D = A(16x4) * B(4x16) + C(16x16), all f32. Per-lane: A,B = float2, C/D = float8.
static __device__ __forceinline__ v8f wmma4(v2f a, v2f b, v8f c) {
    return __builtin_amdgcn_wmma_f32_16x16x4_f32(false, a, false, b, (short)0, c, false, false);
}

// Async copy engine: 16B per lane, global -> LDS, tracked on ASYNCcnt.
static __device__ __forceinline__ void async_b128(const float* g, const float* l) {
    unsigned loff = (unsigned)(uintptr_t)l;   // low 32 bits of LDS aperture addr = LDS offset
    asm volatile("global_load_async_to_lds_b128 %0, %1, off"
                 :: "v"(loff), "v"(g) : "memory");
}
static __device__ __forceinline__ void wait_async16() {
    asm volatile("s_wait_asynccnt 0x10" ::: "memory");
}
static __device__ __forceinline__ void wait_async0() {
    asm volatile("s_wait_asynccnt 0x0" ::: "memory");
}
static __device__ __forceinline__ void wait_ds0() {
    asm volatile("s_wait_dscnt 0x0" ::: "memory");
}

// ---------------------------------------------------------------------------
// Kernel 1: fused QKV projection.  C[m,n] = x[m,:] . W[n,:] + b[n]
// grid: (M/16, N/64, 3), block: 32 (one wave -> 16x64 output tile)
// ---------------------------------------------------------------------------
__global__ void qkv_proj_kernel(const float* __restrict__ x,
                                const float* __restrict__ Wq, const float* __restrict__ bq,
                                const float* __restrict__ Wk, const float* __restrict__ bk,
                                const float* __restrict__ Wv, const float* __restrict__ bv,
                                float* __restrict__ Q, float* __restrict__ K,
                                float* __restrict__ V) {
    const int lane  = threadIdx.x;
    const int m0    = blockIdx.x * 16;
    const int n0    = blockIdx.y * 64;
    const int which = blockIdx.z;
    const float* W    = (which == 0) ? Wq : (which == 1) ? Wk : Wv;
    const float* bias = (which == 0) ? bq : (which == 1) ? bk : bv;
    float*       dst  = (which == 0) ? Q  : (which == 1) ? K  : V;

    const int nlane = lane & 15;
    const int khalf = (lane >> 4) << 1;
    const int rhalf = (lane >> 4) * 8;

    const float* arow = x + (size_t)(m0 + nlane) * EMBED + khalf;
    const float* wrow = W + (size_t)(n0 + nlane) * EMBED + khalf;

    v8f acc0 = {}, acc1 = {}, acc2 = {}, acc3 = {};
    #pragma unroll 4
    for (int k = 0; k < EMBED; k += 4) {
        v2f a  = *(const v2f*)(arow + k);
        v2f b0 = *(const v2f*)(wrow + k);
        v2f b1 = *(const v2f*)(wrow + 16 * EMBED + k);
        v2f b2 = *(const v2f*)(wrow + 32 * EMBED + k);
        v2f b3 = *(const v2f*)(wrow + 48 * EMBED + k);
        acc0 = wmma4(a, b0, acc0);
        acc1 = wmma4(a, b1, acc1);
        acc2 = wmma4(a, b2, acc2);
        acc3 = wmma4(a, b3, acc3);
    }

    #pragma unroll
    for (int t = 0; t < 4; ++t) {
        v8f acc = (t == 0) ? acc0 : (t == 1) ? acc1 : (t == 2) ? acc2 : acc3;
        const int   col = n0 + t * 16 + nlane;
        const float bb  = bias[col];
        const int   h   = col >> 6;
        const int   d   = col & 63;
        #pragma unroll
        for (int r = 0; r < 8; ++r) {
            const int row  = m0 + r + rhalf;
            const int bidx = row >> 11;
            const int s    = row & (SEQ - 1);
            dst[(((size_t)bidx * HEADS + h) * SEQ + s) * HDIM + d] = acc[r] + bb;
        }
    }
}

// ---------------------------------------------------------------------------
// Kernel 2: flash-attention, prefix-causal + padding mask.
// grid: (S/16, B*H), block: 32.  K/V tiles double-buffered in LDS via the
// async copy engine (ASYNCcnt software pipeline); P staged through LDS for
// the C->A layout swap; online softmax via 16-lane xor-butterflies.
// ---------------------------------------------------------------------------
__global__ void attn_kernel(const float* __restrict__ Q, const float* __restrict__ K,
                            const float* __restrict__ V, const int* __restrict__ pad,
                            float* __restrict__ ctx) {
    const int lane = threadIdx.x;
    const int qt   = blockIdx.x;              // 0..127
    const int bh   = blockIdx.y;              // 0..47
    const int b    = bh / HEADS;
    const int h    = bh % HEADS;
    const int q0   = qt * 16;
    const size_t base = (size_t)bh * SEQ * HDIM;

    const int nlane = lane & 15;
    const int khalf = (lane >> 4) << 1;
    const int rhalf = (lane >> 4) * 8;

    __shared__ float kbuf[2][16 * KVPAD];
    __shared__ float vbuf[2][16 * KVPAD];
    __shared__ float plds[16 * 16];

    // Issue async copies for one 16-key K+V tile (8+8 x b128, fully coalesced).
    auto issue_tile = [&](int j, int buf) {
        const float* kg = K + base + (size_t)j * 16 * HDIM;
        const float* vg = V + base + (size_t)j * 16 * HDIM;
        int flat = lane * 4;                       // 4 floats per lane per transfer
        #pragma unroll
        for (int i = 0; i < 8; ++i, flat += 128) { // 2 rows per transfer
            const int row = flat >> 6, col = flat & 63;
            async_b128(kg + flat, &kbuf[buf][row * KVPAD + col]);
            async_b128(vg + flat, &vbuf[buf][row * KVPAD + col]);
        }
    };

    // Preload 16 query A-fragments, 1/sqrt(D)=1/8 folded in.
    v2f qf[16];
    const float* qrow = Q + base + (size_t)(q0 + nlane) * HDIM + khalf;
    #pragma unroll
    for (int kk = 0; kk < 16; ++kk) {
        v2f t = *(const v2f*)(qrow + kk * 4);
        qf[kk] = t * 0.125f;
    }

    v8f o0 = {}, o1 = {}, o2 = {}, o3 = {};
    float mstate[8], lstate[8];
    #pragma unroll
    for (int r = 0; r < 8; ++r) { mstate[r] = -INFINITY; lstate[r] = 0.f; }

    const int jmax = (qt > 12) ? qt : 12;     // (IMG-1)/16 == 12: prefix always visible
    issue_tile(0, 0);

    for (int j = 0; j <= jmax; ++j) {
        const int buf = j & 1;
        if (j < jmax) {
            wait_ds0();                        // don't overwrite a buffer with DS reads in flight
            issue_tile(j + 1, buf ^ 1);
            wait_async16();                    // tile j's 16 transfers complete
        } else {
            wait_async0();
        }

        // S = Q . K^T from LDS-staged K tile (conflict-free ds_load_b64)
        const float* kt = &kbuf[buf][nlane * KVPAD + khalf];
        v8f s = {};
        #pragma unroll
        for (int kk = 0; kk < 16; ++kk) {
            v2f kb = *(const v2f*)(kt + kk * 4);
            s = wmma4(qf[kk], kb, s);
        }

        // prefix-causal + key padding mask (selects only)
        const int  col   = j * 16 + nlane;
        const bool padok = (col < IMG) ? true : (pad[b * TXT + (col - IMG)] != 0);
        #pragma unroll
        for (int r = 0; r < 8; ++r) {
            const int  row = q0 + r + rhalf;
            const bool ok  = padok && ((col < IMG) || (col <= row));
            if (!ok) s[r] = NEGV;
        }

        // online softmax: rows live inside 16-lane half-waves
        float fac[8];
        #pragma unroll
        for (int r = 0; r < 8; ++r) {
            float v = s[r];
            v = fmaxf(v, __shfl_xor(v, 1, 32));
            v = fmaxf(v, __shfl_xor(v, 2, 32));
            v = fmaxf(v, __shfl_xor(v, 4, 32));
            v = fmaxf(v, __shfl_xor(v, 8, 32));
            const float mnew = fmaxf(mstate[r], v);
            const float p    = __expf(s[r] - mnew);
            s[r] = p;
            float rs = p;
            rs += __shfl_xor(rs, 1, 32);
            rs += __shfl_xor(rs, 2, 32);
            rs += __shfl_xor(rs, 4, 32);
            rs += __shfl_xor(rs, 8, 32);
            const float f = __expf(mstate[r] - mnew);
            lstate[r] = lstate[r] * f + rs;
            mstate[r] = mnew;
            fac[r]    = f;
        }
        #pragma unroll
        for (int r = 0; r < 8; ++r) {
            o0[r] *= fac[r]; o1[r] *= fac[r]; o2[r] *= fac[r]; o3[r] *= fac[r];
        }

        // C-layout -> A-layout for P via LDS
        __syncthreads();
        #pragma unroll
        for (int r = 0; r < 8; ++r)
            plds[(r + rhalf) * 16 + nlane] = s[r];
        __syncthreads();

        // O += P . V from LDS-staged V tile
        const float* vt = vbuf[buf];
        #pragma unroll
        for (int st = 0; st < 4; ++st) {
            v2f p2 = *(const v2f*)(&plds[nlane * 16 + st * 4 + khalf]);
            const int k0 = st * 4 + khalf;
            const int dc = nlane;
            #pragma unroll
            for (int t = 0; t < 4; ++t) {
                v2f bf;
                bf.x = vt[k0 * KVPAD + t * 16 + dc];
                bf.y = vt[(k0 + 1) * KVPAD + t * 16 + dc];
                v8f& oo = (t == 0) ? o0 : (t == 1) ? o1 : (t == 2) ? o2 : o3;
                oo = wmma4(p2, bf, oo);
            }
        }
    }

    // normalize + write context in [B,S,E]
    float inv[8];
    #pragma unroll
    for (int r = 0; r < 8; ++r) inv[r] = 1.f / lstate[r];
    #pragma unroll
    for (int t = 0; t < 4; ++t) {
        v8f oo = (t == 0) ? o0 : (t == 1) ? o1 : (t == 2) ? o2 : o3;
        const int e = h * HDIM + t * 16 + nlane;
        #pragma unroll
        for (int r = 0; r < 8; ++r) {
            const int srow = q0 + r + rhalf;
            ctx[((size_t)b * SEQ + srow) * EMBED + e] = oo[r] * inv[r];
        }
    }
}

// ---------------------------------------------------------------------------
// Kernel 3: output projection.  out[m,n] = ctx[m,:] . Wo[n,:] + bo[n]
// ---------------------------------------------------------------------------
__global__ void out_proj_kernel(const float* __restrict__ ctx,
                                const float* __restrict__ Wo,
                                const float* __restrict__ bo,
                                float* __restrict__ out) {
    const int lane  = threadIdx.x;
    const int m0    = blockIdx.x * 16;
    const int n0    = blockIdx.y * 64;
    const int nlane = lane & 15;
    const int khalf = (lane >> 4) << 1;
    const int rhalf = (lane >> 4) * 8;

    const float* arow = ctx + (size_t)(m0 + nlane) * EMBED + khalf;
    const float* wrow = Wo  + (size_t)(n0 + nlane) * EMBED + khalf;

    v8f acc0 = {}, acc1 = {}, acc2 = {}, acc3 = {};
    #pragma unroll 4
    for (int k = 0; k < EMBED; k += 4) {
        v2f a  = *(const v2f*)(arow + k);
        v2f b0 = *(const v2f*)(wrow + k);
        v2f b1 = *(const v2f*)(wrow + 16 * EMBED + k);
        v2f b2 = *(const v2f*)(wrow + 32 * EMBED + k);
        v2f b3 = *(const v2f*)(wrow + 48 * EMBED + k);
        acc0 = wmma4(a, b0, acc0);
        acc1 = wmma4(a, b1, acc1);
        acc2 = wmma4(a, b2, acc2);
        acc3 = wmma4(a, b3, acc3);
    }

    #pragma unroll
    for (int t = 0; t < 4; ++t) {
        v8f acc = (t == 0) ? acc0 : (t == 1) ? acc1 : (t == 2) ? acc2 : acc3;
        const int   col = n0 + t * 16 + nlane;
        const float bb  = bo[col];
        #pragma unroll
        for (int r = 0; r < 8; ++r) {
            const int row = m0 + r + rhalf;
            out[(size_t)row * EMBED + col] = acc[r] + bb;
        }
    }
}

// ---------------------------------------------------------------------------
extern "C" void kernel_launch(void* const* d_in, const int* in_sizes, int n_in,
                              void* d_out, int out_size, void* d_ws, size_t ws_size,
                              hipStream_t stream) {
    const float* x   = (const float*)d_in[0];
    const int*   pad = (const int*)  d_in[1];
    const float* Wq  = (const float*)d_in[2];
    const float* bq  = (const float*)d_in[3];
    const float* Wk  = (const float*)d_in[4];
    const float* bk  = (const float*)d_in[5];
    const float* Wv  = (const float*)d_in[6];
    const float* bv  = (const float*)d_in[7];
    const float* Wo  = (const float*)d_in[8];
    const float* bo  = (const float*)d_in[9];
    float*       out = (float*)d_out;

    // Workspace: Q, K, V in [B,H,S,D] + ctx in [B,S,E]; 100,663,296 bytes.
    const size_t NELT = (size_t)BATCHN * SEQ * EMBED;   // 6,291,456
    float* Q   = (float*)d_ws;
    float* K   = Q + NELT;
    float* V   = K + NELT;
    float* ctx = V + NELT;

    const dim3 blk(32);
    const int MT = (BATCHN * SEQ) / 16;   // 512
    const int NT = EMBED / 64;            // 12

    qkv_proj_kernel<<<dim3(MT, NT, 3), blk, 0, stream>>>(x, Wq, bq, Wk, bk, Wv, bv, Q, K, V);
    attn_kernel<<<dim3(SEQ / 16, BATCHN * HEADS), blk, 0, stream>>>(Q, K, V, pad, ctx);
    out_proj_kernel<<<dim3(MT, NT), blk, 0, stream>>>(ctx, Wo, bo, out);
}